// OffsetBlock_3865470567125
// MI455X (gfx1250) — compile-verified
//
#include <hip/hip_runtime.h>

// Paired-corner 64-bit load with only 4-byte alignment guaranteed
// (xa may be odd). aligned(4) on the typedef lowers the assumed alignment.
typedef float vf2 __attribute__((ext_vector_type(2)));
typedef vf2 __attribute__((aligned(4))) vf2_a4;

namespace {
constexpr int N  = 8;
constexpr int C  = 256;
constexpr int H  = 128;
constexpr int W  = 128;
constexpr int HW = H * W;
constexpr int CPB = 32;  // channels per block; gridDim.y = C / CPB
}

__global__ __launch_bounds__(256)
void offset_sample_kernel(const float* __restrict__ xin,
                          const float* __restrict__ offmap,
                          float* __restrict__ out) {
  const int t  = blockIdx.x * 256 + threadIdx.x;  // pixel id in [0, N*H*W)
  const int xq = t & (W - 1);
  const int yq = (t >> 7) & (H - 1);
  const int n  = t >> 14;
  const int pix = yq * W + xq;

  // offset_map: (N, 2, H, W); channel 0 = h-offset, channel 1 = w-offset
  const size_t obase = (size_t)n * 2 * HW + (size_t)pix;
  const float offh = offmap[obase];
  const float offw = offmap[obase + HW];

  // Replicate the reference arithmetic sequence exactly.
  const float grid_h = offh + (float)yq;
  const float grid_w = offw + (float)xq;
  const float gx = grid_w / ((float)(W - 1) * 0.5f) - 1.0f;
  const float gy = grid_h / ((float)(H - 1) * 0.5f) - 1.0f;
  float ix = ((gx + 1.0f) * (float)W - 1.0f) * 0.5f;
  float iy = ((gy + 1.0f) * (float)H - 1.0f) * 0.5f;
  // border padding: clamp BEFORE computing corners/weights
  ix = fminf(fmaxf(ix, 0.0f), (float)(W - 1));
  iy = fminf(fmaxf(iy, 0.0f), (float)(H - 1));

  const int x0 = (int)floorf(ix);
  const int y0 = (int)floorf(iy);
  // Shifted base: if x0==W-1 then ix==W-1 exactly (post-clamp), so using
  // xa=W-2 with fx=ix-xa=1.0 reproduces the clamped-corner result exactly,
  // and the two horizontal corners are always adjacent -> one b64 load.
  const int xa = min(x0, W - 2);
  const int ya = min(y0, H - 2);
  const float fx = ix - (float)xa;
  const float fy = iy - (float)ya;
  const float wx0 = 1.0f - fx, wx1 = fx;
  const float wy0 = 1.0f - fy, wy1 = fy;

  const int oT = ya * W + xa;   // top corner pair
  const int oB = oT + W;        // bottom corner pair

  const int c0 = blockIdx.y * CPB;
  const float* __restrict__ p = xin + ((size_t)n * C + c0) * (size_t)HW;
  float* __restrict__ q       = out + ((size_t)n * C + c0) * (size_t)HW + pix;

#pragma unroll 8
  for (int c = 0; c < CPB; ++c) {
    const vf2 tv = *reinterpret_cast<const vf2_a4*>(p + oT);
    const vf2 bv = *reinterpret_cast<const vf2_a4*>(p + oB);
    const float top = tv.x * wx0 + tv.y * wx1;
    const float bot = bv.x * wx0 + bv.y * wx1;
    const float r   = top * wy0 + bot * wy1;
    // Output is write-once/never re-read: non-temporal store keeps the
    // 192MB L2 dedicated to x (which fits entirely), so the 4-corner
    // gather re-reads never go back to HBM.
    __builtin_nontemporal_store(r, q);
    p += HW;
    q += HW;
  }
}

extern "C" void kernel_launch(void* const* d_in, const int* in_sizes, int n_in,
                              void* d_out, int out_size, void* d_ws, size_t ws_size,
                              hipStream_t stream) {
  const float* x   = (const float*)d_in[0];  // (8, 256, 128, 128) f32
  const float* off = (const float*)d_in[1];  // (8, 2, 128, 128) f32
  float* out = (float*)d_out;                // (8, 256, 128, 128) f32

  dim3 grid((N * H * W) / 256, C / CPB, 1);  // 512 x 8 blocks
  dim3 block(256, 1, 1);                     // 8 wave32s per block
  hipLaunchKernelGGL(offset_sample_kernel, grid, block, 0, stream, x, off, out);
}